// SimOTA_22522808500278
// MI455X (gfx1250) — compile-verified
//
#include <hip/hip_runtime.h>

// ---------------- problem constants (from reference) ----------------
#define NUM_CLASSES 80
#define A_N 33600          // anchors (multiple of 16 and of 32)
#define G_N 256            // gts (multiple of 16)
#define TOPK 10
#define RADIUS 2.5f
#define IOU_W 3.0f
#define CLS_W 1.0f
#define BIG_COST 1e10f

typedef __attribute__((ext_vector_type(2))) float v2f;
typedef __attribute__((ext_vector_type(8))) float v8f;

__device__ __forceinline__ float softplus_f(float x) {
  // log(1 + e^x), numerically stable
  return fmaxf(x, 0.0f) + log1pf(expf(-fabsf(x)));
}

// -------------------------------------------------------------------
// Kernel 1: per-anchor precompute.
//   Ssum[a]  = sum_c softplus(scores[a,c])   (== -sum_c log_sigmoid(-x))
//   areaP[a] = pred box area
//   assigned[a] = -1   (re-initialized every call: deterministic)
// -------------------------------------------------------------------
__global__ __launch_bounds__(256) void precompute_kernel(
    const float* __restrict__ scores, const float* __restrict__ pb,
    float* __restrict__ Ssum, float* __restrict__ areaP,
    int* __restrict__ assigned) {
  int a = blockIdx.x * blockDim.x + threadIdx.x;
  if (a >= A_N) return;
  const float4* row = (const float4*)(scores + (size_t)a * NUM_CLASSES);
  float s = 0.0f;
#pragma unroll
  for (int i = 0; i < NUM_CLASSES / 4; ++i) {
    float4 v = row[i];
    s += softplus_f(v.x) + softplus_f(v.y) + softplus_f(v.z) + softplus_f(v.w);
  }
  Ssum[a] = s;
  float4 b = ((const float4*)pb)[a];
  areaP[a] = (b.z - b.x) * (b.w - b.y);
  assigned[a] = -1;
}

// -------------------------------------------------------------------
// Kernel 2: Gathered[g,a] = scores[a, label[g]] as a one-hot GEMM on the
// WMMA f32 path:  Gathered = OneHot(GxK) * Scores^T(KxA), K = 80 classes.
// V_WMMA_F32_16X16X4_F32, exact in f32 (one-hot entries are 0/1).
// One wave per 16x16 output tile, 20 K-steps of 4.
// ISA 7.12.2 f32 layouts:
//   A 16x4 : lanes 0-15 -> M=lane, VGPR0=K0,VGPR1=K1 ; lanes 16-31 -> K2,K3
//   B 4x16 : lanes hold N=lane&15, same K split as A (row-striped)
//   C/D    : VGPR v -> M = v + 8*(lane>=16), N = lane&15
// -------------------------------------------------------------------
__global__ __launch_bounds__(256) void gather_wmma_kernel(
    const float* __restrict__ scores, const int* __restrict__ labels,
    float* __restrict__ gathered) {
  const int lane = threadIdx.x & 31;
  const int wave = blockIdx.x * (blockDim.x >> 5) + (threadIdx.x >> 5);
  const int ablk = wave % (A_N / 16);
  const int gblk = wave / (A_N / 16);
  const int g0 = gblk * 16;
  const int a0 = ablk * 16;
  const int m = lane & 15;
  const int khalf = (lane >> 4) << 1;  // 0 for lanes 0-15, 2 for lanes 16-31

  const int lbl = labels[g0 + m];                              // A-frag row M
  const float* brow = scores + (size_t)(a0 + m) * NUM_CLASSES; // B-frag col N

  v8f acc = {};
#pragma unroll
  for (int k = 0; k < NUM_CLASSES; k += 4) {
    const int c0 = k + khalf;
    const int c1 = c0 + 1;
    v2f av, bv;
    av.x = (lbl == c0) ? 1.0f : 0.0f;   // OneHot[M][c0]
    av.y = (lbl == c1) ? 1.0f : 0.0f;   // OneHot[M][c1]
    bv.x = brow[c0];                    // Scores^T[c0][N]
    bv.y = brow[c1];                    // Scores^T[c1][N]
    acc = __builtin_amdgcn_wmma_f32_16x16x4_f32(
        /*neg_a=*/false, av, /*neg_b=*/false, bv,
        /*c_mod=*/(short)0, acc, /*reuse_a=*/false, /*reuse_b=*/false);
  }

  const int gg = g0 + ((lane >> 4) << 3);  // +8 rows for upper half-wave
  float* out = gathered + (size_t)gg * A_N + a0 + m;
#pragma unroll
  for (int v = 0; v < 8; ++v) out[(size_t)v * A_N] = acc[v];
}

// -------------------------------------------------------------------
// Kernel 3: per-GT cost + streaming top-10 + candidate count.
// One wave32 per GT (8 GTs per 256-thread block). Each lane scans a
// 32-strided slice (exactly 1050 anchors), keeps a sorted top-10 in
// registers, then a 10-round wave arg-min merge via __shfl_xor.
// -------------------------------------------------------------------
__global__ __launch_bounds__(256) void cost_topk_kernel(
    const float* __restrict__ gathered, const float* __restrict__ Ssum,
    const float* __restrict__ areaP, const float* __restrict__ pb,
    const float* __restrict__ ap, const float* __restrict__ gtb,
    int* __restrict__ topk, int* __restrict__ dynk) {
  const int lane = threadIdx.x & 31;
  const int g = blockIdx.x * (blockDim.x >> 5) + (threadIdx.x >> 5);
  if (g >= G_N) return;

  const float4 gb = ((const float4*)gtb)[g];
  const float areaG = (gb.z - gb.x) * (gb.w - gb.y);
  const float cx = 0.5f * (gb.x + gb.z), cy = 0.5f * (gb.y + gb.w);
  const float rx = RADIUS * (gb.z - gb.x), ry = RADIUS * (gb.w - gb.y);
  const float* grow = gathered + (size_t)g * A_N;

  float tc[TOPK];
  int ti[TOPK];
#pragma unroll
  for (int j = 0; j < TOPK; ++j) { tc[j] = INFINITY; ti[j] = 0; }
  int cnt = 0;

  for (int a = lane; a < A_N; a += 32) {
    const float4 p = ((const float4*)pb)[a];
    const float2 pt = ((const float2*)ap)[a];
    const bool in_gt = (pt.x >= gb.x) && (pt.x <= gb.z) &&
                       (pt.y >= gb.y) && (pt.y <= gb.w);
    const bool in_c = (pt.x >= cx - rx) && (pt.x <= cx + rx) &&
                      (pt.y >= cy - ry) && (pt.y <= cy + ry);
    const bool inside = in_gt && in_c;

    const float ltx = fmaxf(gb.x, p.x), lty = fmaxf(gb.y, p.y);
    const float rbx = fminf(gb.z, p.z), rby = fminf(gb.w, p.w);
    const float w = fmaxf(rbx - ltx, 0.0f), h = fmaxf(rby - lty, 0.0f);
    const float ov = w * h;
    const float iou = ov / (areaG + areaP[a] - ov + 1e-6f);

    cnt += (inside && iou > 0.0f) ? 1 : 0;

    const float cls_cost = Ssum[a] - grow[a];   // softplus-sum - score[label]
    float cost = CLS_W * cls_cost + IOU_W * (-logf(iou));
    if (!inside) cost += BIG_COST;              // iou==0 -> +inf, stays +inf

    if (cost < tc[TOPK - 1]) {
      float cc = cost;
      int ci = a;
#pragma unroll
      for (int j = 0; j < TOPK; ++j) {
        if (cc < tc[j]) {
          float t = tc[j]; tc[j] = cc; cc = t;
          int u = ti[j];  ti[j] = ci; ci = u;
        }
      }
    }
  }

  // wave-wide candidate count
#pragma unroll
  for (int off = 16; off > 0; off >>= 1) cnt += __shfl_xor(cnt, off, 32);

  // 10-round wave arg-min merge of the 32 sorted per-lane lists
  int head = 0;
  for (int k = 0; k < TOPK; ++k) {
    float bc = (head < TOPK) ? tc[head] : INFINITY;
    int bi = (head < TOPK) ? ti[head] : 0;
    int bl = lane;
#pragma unroll
    for (int off = 16; off > 0; off >>= 1) {
      float oc = __shfl_xor(bc, off, 32);
      int oi = __shfl_xor(bi, off, 32);
      int ol = __shfl_xor(bl, off, 32);
      if (oc < bc || (oc == bc && ol < bl)) { bc = oc; bi = oi; bl = ol; }
    }
    if (bl == lane) head++;
    if (lane == 0) topk[g * TOPK + k] = bi;
  }
  if (lane == 0) dynk[g] = min(max(cnt, 1), TOPK);
}

// -------------------------------------------------------------------
// Kernel 4: scatter-max GT ids onto anchors (max is order-independent,
// so atomics keep this deterministic across replays).
// -------------------------------------------------------------------
__global__ __launch_bounds__(256) void scatter_kernel(
    const int* __restrict__ topk, const int* __restrict__ dynk,
    int* __restrict__ assigned) {
  int g = blockIdx.x * blockDim.x + threadIdx.x;
  if (g >= G_N) return;
  int k = dynk[g];
  for (int j = 0; j < k; ++j) atomicMax(&assigned[topk[g * TOPK + j]], g);
}

// -------------------------------------------------------------------
// Kernel 5: finalize per anchor: labels (as f32), boxes, 81-wide scores.
// Fully overwrites d_out every call (harness poisons it with 0xAA).
// -------------------------------------------------------------------
__global__ __launch_bounds__(256) void finalize_kernel(
    const float* __restrict__ pb, const float* __restrict__ gtb,
    const int* __restrict__ glab, const float* __restrict__ areaP,
    const int* __restrict__ assigned, float* __restrict__ out) {
  int a = blockIdx.x * blockDim.x + threadIdx.x;
  if (a >= A_N) return;
  const int g = assigned[a];
  const bool pos = (g >= 0);
  const int sg = pos ? g : 0;
  const int lbl = pos ? glab[sg] : NUM_CLASSES;

  const float4 gb = ((const float4*)gtb)[sg];
  const float4 p = ((const float4*)pb)[a];
  const float ltx = fmaxf(gb.x, p.x), lty = fmaxf(gb.y, p.y);
  const float rbx = fminf(gb.z, p.z), rby = fminf(gb.w, p.w);
  const float w = fmaxf(rbx - ltx, 0.0f), h = fmaxf(rby - lty, 0.0f);
  const float ov = w * h;
  const float areaG = (gb.z - gb.x) * (gb.w - gb.y);
  const float iou = ov / (areaG + areaP[a] - ov + 1e-6f);

  out[a] = (float)lbl;  // assigned_labels as f32

  float* ob = out + A_N + (size_t)a * 4;  // assigned_bboxes
  ob[0] = pos ? gb.x : 0.0f;
  ob[1] = pos ? gb.y : 0.0f;
  ob[2] = pos ? gb.z : 0.0f;
  ob[3] = pos ? gb.w : 0.0f;

  float* os = out + (size_t)A_N * 5 + (size_t)a * (NUM_CLASSES + 1);
#pragma unroll
  for (int c = 0; c <= NUM_CLASSES; ++c) os[c] = 0.0f;
  os[lbl] = pos ? iou : 0.0f;  // assigned_scores one-hot IoU
}

// -------------------------------------------------------------------
extern "C" void kernel_launch(void* const* d_in, const int* in_sizes, int n_in,
                              void* d_out, int out_size, void* d_ws, size_t ws_size,
                              hipStream_t stream) {
  const float* scores = (const float*)d_in[0];  // (A, 80) f32
  const float* pb     = (const float*)d_in[1];  // (A, 4)  f32
  const float* ap     = (const float*)d_in[2];  // (A, 2)  f32
  const int*   glab   = (const int*)d_in[3];    // (G,)    i32
  const float* gtb    = (const float*)d_in[4];  // (G, 4)  f32
  float* out = (float*)d_out;

  // workspace layout (~34.8 MB)
  float* gathered = (float*)d_ws;                      // G*A
  float* Ssum     = gathered + (size_t)G_N * A_N;      // A
  float* areaP    = Ssum + A_N;                        // A
  int*   topk     = (int*)(areaP + A_N);               // G*TOPK
  int*   dynk     = topk + G_N * TOPK;                 // G
  int*   assigned = dynk + G_N;                        // A

  precompute_kernel<<<(A_N + 255) / 256, 256, 0, stream>>>(
      scores, pb, Ssum, areaP, assigned);

  const int waves = (G_N / 16) * (A_N / 16);  // 33600 tiles, 8 waves/block
  gather_wmma_kernel<<<waves / 8, 256, 0, stream>>>(scores, glab, gathered);

  cost_topk_kernel<<<G_N / 8, 256, 0, stream>>>(
      gathered, Ssum, areaP, pb, ap, gtb, topk, dynk);

  scatter_kernel<<<1, G_N, 0, stream>>>(topk, dynk, assigned);

  finalize_kernel<<<(A_N + 255) / 256, 256, 0, stream>>>(
      pb, gtb, glab, areaP, assigned, out);
}